// HamiltonianEvolution_54494545051761
// MI455X (gfx1250) — compile-verified
//
#include <hip/hip_runtime.h>
#include <hip/hip_bf16.h>

typedef float v2f __attribute__((ext_vector_type(2)));
typedef float v8f __attribute__((ext_vector_type(8)));

constexpr int ND = 64;         // phase-space dim
constexpr int NH = 128;        // hidden dim
constexpr int NT = 64;         // leapfrog steps
constexpr int BT = 16;         // batch rows per wave (= WMMA N)
constexpr int NWAVES = 4;
constexpr int BLOCK = NWAVES * 32;
constexpr int NB = 4096;

// All LDS row strides chosen ≡ 20 (mod 64): conflict-free for A-frag (lane=row),
// transposed-A, B-frag, and C-tile store access patterns of f32 16x16x4 WMMA.
constexpr int SW1 = ND + 20;   // 84
constexpr int SW2 = NH + 20;   // 148
constexpr int SMI = ND + 20;   // 84
constexpr int SAC = BT + 4;    // 20 (activation buffers: [feature][batch16])

constexpr float DT  = 1.0f / NT;
constexpr float HDT = 0.5f * DT;

// ---------------- f32 WMMA 16x16x4 fragments ----------------
// A (16x4): lane L: row = L%16; vgpr0 = K(2*(L/16)), vgpr1 = K(2*(L/16)+1)
// B (4x16): lane L: col = L%16; vgpr0 = K(2*(L/16)) row, vgpr1 = next K row
// C (16x16): vgpr m, lane L: row = m + 8*(L/16), col = L%16

__device__ __forceinline__ v2f frag_a(const float* base, int stride, int m0, int k0, int r, int half) {
  const float* p = base + (m0 + r) * stride + (k0 + 2 * half);
  v2f a; a.x = p[0]; a.y = p[1]; return a;
}
// logical A[m][k] = base[k*stride + m]  (transposed weight access)
__device__ __forceinline__ v2f frag_at(const float* base, int stride, int m0, int k0, int r, int half) {
  const float* p = base + (k0 + 2 * half) * stride + (m0 + r);
  v2f a; a.x = p[0]; a.y = p[stride]; return a;
}
__device__ __forceinline__ v2f frag_b(const float* base, int stride, int k0, int n0, int r, int half) {
  const float* p = base + (k0 + 2 * half) * stride + (n0 + r);
  v2f b; b.x = p[0]; b.y = p[stride]; return b;
}
__device__ __forceinline__ v8f wmma4(v2f a, v2f b, v8f c) {
  return __builtin_amdgcn_wmma_f32_16x16x4_f32(false, a, false, b, (short)0, c, false, false);
}

// sigmoid via v_exp + v_rcp (avoids the IEEE v_div_scale/newton sequence)
__device__ __forceinline__ float fast_sigmoid(float x) {
  return __builtin_amdgcn_rcpf(1.0f + __expf(-x));
}

// epilogue helpers (operate on one 16x16 C tile at feature-row block j0)
__device__ __forceinline__ void silu_epi(v8f c, int j0, int r, int half, float* hT, float* gT) {
  #pragma unroll
  for (int m = 0; m < 8; ++m) {
    const int row = j0 + m + 8 * half;
    float x = c[m];
    float s = fast_sigmoid(x);
    hT[row * SAC + r] = x * s;
    gT[row * SAC + r] = s * (1.0f + x * (1.0f - s));
  }
}
__device__ __forceinline__ void dsilu_w3_epi(v8f c, int j0, int r, int half,
                                             const float* sw3, float* sT) {
  #pragma unroll
  for (int m = 0; m < 8; ++m) {
    const int row = j0 + m + 8 * half;
    float x = c[m];
    float s = fast_sigmoid(x);
    sT[row * SAC + r] = s * (1.0f + x * (1.0f - s)) * sw3[row];
  }
}
__device__ __forceinline__ void mulg_epi(v8f c, int j0, int r, int half, float* gT) {
  #pragma unroll
  for (int m = 0; m < 8; ++m) {
    const int idx = (j0 + m + 8 * half) * SAC + r;
    gT[idx] = c[m] * gT[idx];
  }
}

// gradV(q) through the MLP (transposed GEMMs, batch = N dim), fused with the
// momentum half-kick(s). FIRST: p -= 0.5*dt*g (initial half-kick only).
// Otherwise: p_new = p_half - 0.5*dt*g staged into sT (for trajectory), and
// pT = p_new - 0.5*dt*g = p_half of the NEXT step (gradV reuse across steps).
// All GEMMs process two M-tiles per pass: two independent WMMA accumulator
// chains sharing one B fragment (2x ILP, half the B-side LDS traffic).
template <bool FIRST>
__device__ __forceinline__ void grad_and_kick(
    const float* sW1, const float* sW2,
    const float* sb1, const float* sb2, const float* sw3,
    const float* qT, float* pT, float* hT, float* gT, float* sT,
    int r, int half)
{
  // a1^T = W1 @ q^T + b1 ;  hT = silu(a1), gT = silu'(a1)
  for (int jt = 0; jt < NH / 16; jt += 2) {
    const int j0 = jt * 16, j1 = j0 + 16;
    v8f c0, c1;
    #pragma unroll
    for (int m = 0; m < 8; ++m) { c0[m] = sb1[j0 + m + 8 * half]; c1[m] = sb1[j1 + m + 8 * half]; }
    #pragma unroll
    for (int k = 0; k < ND; k += 4) {
      v2f b = frag_b(qT, SAC, k, 0, r, half);
      c0 = wmma4(frag_a(sW1, SW1, j0, k, r, half), b, c0);
      c1 = wmma4(frag_a(sW1, SW1, j1, k, r, half), b, c1);
    }
    silu_epi(c0, j0, r, half, hT, gT);
    silu_epi(c1, j1, r, half, hT, gT);
  }
  // a2^T = W2 @ h^T + b2 ;  sT = silu'(a2) * w3
  for (int jt = 0; jt < NH / 16; jt += 2) {
    const int j0 = jt * 16, j1 = j0 + 16;
    v8f c0, c1;
    #pragma unroll
    for (int m = 0; m < 8; ++m) { c0[m] = sb2[j0 + m + 8 * half]; c1[m] = sb2[j1 + m + 8 * half]; }
    #pragma unroll
    for (int k = 0; k < NH; k += 4) {
      v2f b = frag_b(hT, SAC, k, 0, r, half);
      c0 = wmma4(frag_a(sW2, SW2, j0, k, r, half), b, c0);
      c1 = wmma4(frag_a(sW2, SW2, j1, k, r, half), b, c1);
    }
    dsilu_w3_epi(c0, j0, r, half, sw3, sT);
    dsilu_w3_epi(c1, j1, r, half, sw3, sT);
  }
  // d^T = (W2^T @ s^T) * silu'(a1)  -> overwrite gT (element-exact reuse)
  for (int jt = 0; jt < NH / 16; jt += 2) {
    const int j0 = jt * 16, j1 = j0 + 16;
    v8f c0 = {}, c1 = {};
    #pragma unroll
    for (int k = 0; k < NH; k += 4) {
      v2f b = frag_b(sT, SAC, k, 0, r, half);
      c0 = wmma4(frag_at(sW2, SW2, j0, k, r, half), b, c0);
      c1 = wmma4(frag_at(sW2, SW2, j1, k, r, half), b, c1);
    }
    mulg_epi(c0, j0, r, half, gT);
    mulg_epi(c1, j1, r, half, gT);
  }
  // g_q^T = W1^T @ d^T ; kick p
  for (int it = 0; it < ND / 16; it += 2) {
    const int d0 = it * 16, d1 = d0 + 16;
    v8f c0 = {}, c1 = {};
    #pragma unroll
    for (int k = 0; k < NH; k += 4) {
      v2f b = frag_b(gT, SAC, k, 0, r, half);
      c0 = wmma4(frag_at(sW1, SW1, d0, k, r, half), b, c0);
      c1 = wmma4(frag_at(sW1, SW1, d1, k, r, half), b, c1);
    }
    #pragma unroll
    for (int m = 0; m < 8; ++m) {
      const int i0 = (d0 + m + 8 * half) * SAC + r;
      const int i1 = (d1 + m + 8 * half) * SAC + r;
      if (FIRST) {
        pT[i0] -= HDT * c0[m];
        pT[i1] -= HDT * c1[m];
      } else {
        float pn0 = pT[i0] - HDT * c0[m];
        float pn1 = pT[i1] - HDT * c1[m];
        sT[i0] = pn0; pT[i0] = pn0 - HDT * c0[m];
        sT[i1] = pn1; pT[i1] = pn1 - HDT * c1[m];
      }
    }
  }
}

// ---------------- kernel 1: Mi = dt * (L L^T + 0.01 I)^-1  (Gauss-Jordan) ----------------
__global__ __launch_bounds__(256) void invert_mass_kernel(const float* __restrict__ L,
                                                          float* __restrict__ Mi) {
  __shared__ float A[ND][2 * ND];
  __shared__ float colk[ND];
  const int tid = threadIdx.x;
  for (int idx = tid; idx < ND * ND; idx += 256) {
    const int rr = idx >> 6, cc = idx & 63;
    float s = (rr == cc) ? 0.01f : 0.0f;
    for (int k = 0; k < ND; ++k) s += L[rr * ND + k] * L[cc * ND + k];
    A[rr][cc] = s;
    A[rr][ND + cc] = (rr == cc) ? 1.0f : 0.0f;
  }
  __syncthreads();
  for (int k = 0; k < ND; ++k) {
    const float inv = 1.0f / A[k][k];   // exact IEEE divide: accuracy matters here
    __syncthreads();
    if (tid < 2 * ND) A[k][tid] *= inv;
    __syncthreads();
    if (tid < ND) colk[tid] = A[tid][k];
    __syncthreads();
    for (int idx = tid; idx < ND * 2 * ND; idx += 256) {
      const int rr = idx >> 7, cc = idx & 127;
      if (rr != k) A[rr][cc] -= colk[rr] * A[k][cc];
    }
    __syncthreads();
  }
  for (int idx = tid; idx < ND * ND; idx += 256)
    Mi[idx] = DT * A[idx >> 6][ND + (idx & 63)];
}

// ---------------- kernel 2: encode z -> (q0,p0) into trajectory slot t=0 ----------------
__global__ __launch_bounds__(ND) void encode_kernel(const float* __restrict__ z,
                                                    const float* __restrict__ qW, const float* __restrict__ qb,
                                                    const float* __restrict__ pW, const float* __restrict__ pb,
                                                    float* __restrict__ out) {
  __shared__ float zs[ND];
  const int b = blockIdx.x;
  const int d = threadIdx.x;
  zs[d] = z[(size_t)b * ND + d];
  __syncthreads();
  float aq = qb[d], ap = pb[d];
  #pragma unroll 8
  for (int e = 0; e < ND; ++e) {
    const float ze = zs[e];
    aq += ze * qW[d * ND + e];
    ap += ze * pW[d * ND + e];
  }
  const size_t o = (size_t)b * (NT + 1) * (2 * ND);
  out[o + d] = aq;
  out[o + ND + d] = ap;
}

// ---------------- kernel 3: full 64-step leapfrog, 16 batch rows per wave ----------------
__global__ __launch_bounds__(BLOCK) void ham_traj_kernel(
    const float* __restrict__ W1, const float* __restrict__ b1,
    const float* __restrict__ W2, const float* __restrict__ b2,
    const float* __restrict__ w3, const float* __restrict__ Mi,
    float* __restrict__ out)
{
  __shared__ float sW1[NH * SW1];
  __shared__ float sW2[NH * SW2];
  __shared__ float sMi[ND * SMI];
  __shared__ float sb1[NH], sb2[NH], sw3[NH];
  __shared__ float shh[NWAVES][NH * SAC];   // h1^T
  __shared__ float shg[NWAVES][NH * SAC];   // silu'(a1) then d^T
  __shared__ float shs[NWAVES][NH * SAC];   // s2^T then p_new staging
  __shared__ float shq[NWAVES][ND * SAC];   // q^T
  __shared__ float shp[NWAVES][ND * SAC];   // p^T (p_half)

  const int tid  = threadIdx.x;
  const int wave = tid >> 5;
  const int lane = tid & 31;
  const int r    = lane & 15;
  const int half = lane >> 4;
  const int bBase = blockIdx.x * (NWAVES * BT) + wave * BT;

  for (int i = tid; i < NH * ND; i += BLOCK) sW1[(i / ND) * SW1 + (i % ND)] = W1[i];
  for (int i = tid; i < NH * NH; i += BLOCK) sW2[(i / NH) * SW2 + (i % NH)] = W2[i];
  for (int i = tid; i < ND * ND; i += BLOCK) sMi[(i / ND) * SMI + (i % ND)] = Mi[i];
  for (int i = tid; i < NH; i += BLOCK) { sb1[i] = b1[i]; sb2[i] = b2[i]; sw3[i] = w3[i]; }
  __syncthreads();

  float* qT = shq[wave]; float* pT = shp[wave];
  float* hT = shh[wave]; float* gT = shg[wave]; float* sT = shs[wave];

  // load (q0,p0) from trajectory slot t=0 (written by encode_kernel)
  for (int i = lane; i < BT * 2 * ND; i += 32) {
    const int b = i >> 7;
    const int f = i & 127;
    float v = out[((size_t)(bBase + b) * (NT + 1)) * (2 * ND) + f];
    if (f < ND) qT[f * SAC + b] = v;
    else        pT[(f - ND) * SAC + b] = v;
  }
  __syncthreads();

  // initial half-kick: p_half_1 = p0 - 0.5*dt*gradV(q0)
  grad_and_kick<true>(sW1, sW2, sb1, sb2, sw3, qT, pT, hT, gT, sT, r, half);

  #pragma unroll 1
  for (int t = 1; t <= NT; ++t) {
    // drift: q^T += Mi @ p_half^T   (Mi = dt * M^-1, WMMA accumulates onto q)
    for (int it = 0; it < ND / 16; it += 2) {
      const int d0 = it * 16, d1 = d0 + 16;
      v8f c0, c1;
      #pragma unroll
      for (int m = 0; m < 8; ++m) {
        c0[m] = qT[(d0 + m + 8 * half) * SAC + r];
        c1[m] = qT[(d1 + m + 8 * half) * SAC + r];
      }
      #pragma unroll
      for (int k = 0; k < ND; k += 4) {
        v2f b = frag_b(pT, SAC, k, 0, r, half);
        c0 = wmma4(frag_a(sMi, SMI, d0, k, r, half), b, c0);
        c1 = wmma4(frag_a(sMi, SMI, d1, k, r, half), b, c1);
      }
      #pragma unroll
      for (int m = 0; m < 8; ++m) {
        qT[(d0 + m + 8 * half) * SAC + r] = c0[m];
        qT[(d1 + m + 8 * half) * SAC + r] = c1[m];
      }
    }
    // trajectory q rows (coalesced float4 via LDS transpose)
    for (int i = lane; i < BT * (ND / 4); i += 32) {
      const int b  = i >> 4;
      const int f4 = (i & 15) * 4;
      float4 v;
      v.x = qT[(f4 + 0) * SAC + b];
      v.y = qT[(f4 + 1) * SAC + b];
      v.z = qT[(f4 + 2) * SAC + b];
      v.w = qT[(f4 + 3) * SAC + b];
      *(float4*)&out[((size_t)(bBase + b) * (NT + 1) + t) * (2 * ND) + f4] = v;
    }
    // gradV(q_new): p_new staged in sT, pT becomes next p_half (grad reuse)
    grad_and_kick<false>(sW1, sW2, sb1, sb2, sw3, qT, pT, hT, gT, sT, r, half);
    // trajectory p rows
    for (int i = lane; i < BT * (ND / 4); i += 32) {
      const int b  = i >> 4;
      const int f4 = (i & 15) * 4;
      float4 v;
      v.x = sT[(f4 + 0) * SAC + b];
      v.y = sT[(f4 + 1) * SAC + b];
      v.z = sT[(f4 + 2) * SAC + b];
      v.w = sT[(f4 + 3) * SAC + b];
      *(float4*)&out[((size_t)(bBase + b) * (NT + 1) + t) * (2 * ND) + ND + f4] = v;
    }
  }
}

extern "C" void kernel_launch(void* const* d_in, const int* in_sizes, int n_in,
                              void* d_out, int out_size, void* d_ws, size_t ws_size,
                              hipStream_t stream) {
  const float* z     = (const float*)d_in[0];
  const float* massL = (const float*)d_in[1];
  const float* Vw1   = (const float*)d_in[2];
  const float* Vb1   = (const float*)d_in[3];
  const float* Vw2   = (const float*)d_in[4];
  const float* Vb2   = (const float*)d_in[5];
  const float* Vw3   = (const float*)d_in[6];
  // d_in[7] = Vb3: constant offset, does not affect gradV -> unused
  const float* qW    = (const float*)d_in[8];
  const float* qb    = (const float*)d_in[9];
  const float* pW    = (const float*)d_in[10];
  const float* pb    = (const float*)d_in[11];
  // d_in[12] = num_steps (== 64, fixed by setup_inputs; out_size matches)
  float* out = (float*)d_out;
  float* Mi  = (float*)d_ws;   // 64*64 f32 scratch

  invert_mass_kernel<<<1, 256, 0, stream>>>(massL, Mi);
  encode_kernel<<<NB, ND, 0, stream>>>(z, qW, qb, pW, pb, out);
  ham_traj_kernel<<<NB / (NWAVES * BT), BLOCK, 0, stream>>>(Vw1, Vb1, Vw2, Vb2, Vw3, Mi, out);
}